// CNNBranch_59596966199731
// MI455X (gfx1250) — compile-verified
//
#include <hip/hip_runtime.h>
#include <hip/hip_bf16.h>

typedef float v2f __attribute__((ext_vector_type(2)));
typedef float v8f __attribute__((ext_vector_type(8)));
typedef int   v4i __attribute__((ext_vector_type(4)));

typedef __attribute__((address_space(1))) v4i* g_v4i_p;   // global
typedef __attribute__((address_space(3))) v4i* l_v4i_p;   // LDS

#define E_DIM 512
#define S_MAX 512
#define T_LEN 65536
#define B_N   128

#define MT  64     // rows per block tile
#define NT  128    // cols per block tile
#define KC  32     // K chunk staged in LDS
#define LDP 36     // LDS row pitch: 16B-aligned rows (144B) + conflict-free frag reads

// ---- async memory->LDS copy of 16 bytes (per lane), ASYNCcnt-tracked -------
__device__ __forceinline__ void async_cp16(const float* g, float* l) {
#if __has_builtin(__builtin_amdgcn_global_load_async_to_lds_b128)
    __builtin_amdgcn_global_load_async_to_lds_b128(
        (g_v4i_p)(v4i*)g, (l_v4i_p)(v4i*)l, 0, 0);
#else
    *(float4*)l = *(const float4*)g;   // sync fallback
#endif
}

__device__ __forceinline__ void wait_async0() {
#if __has_builtin(__builtin_amdgcn_s_wait_asynccnt)
    __builtin_amdgcn_s_wait_asynccnt(0);
#else
    asm volatile("s_wait_asynccnt 0x0" ::: "memory");
#endif
}

// ---------------- Kernel 1: mask popcount -> n_valid + padding_mask output --
__global__ __launch_bounds__(256)
void mask_reduce_kernel(const unsigned char* __restrict__ mask,
                        const int* __restrict__ branch_idx,
                        float* __restrict__ mask_out,
                        int* __restrict__ n_valid_ws) {
    const int b = blockIdx.x;
    const int t = threadIdx.x;
    const uint4* row = (const uint4*)(mask + (size_t)b * T_LEN);
    int sum = 0;
    for (int i = t; i < T_LEN / 16; i += 256) {
        uint4 v = row[i];
        sum += __popc(v.x & 0x01010101u) + __popc(v.y & 0x01010101u)
             + __popc(v.z & 0x01010101u) + __popc(v.w & 0x01010101u);
    }
    __shared__ int red[256];
    __shared__ int nv_sh;
    red[t] = sum;
    __syncthreads();
    for (int s = 128; s > 0; s >>= 1) {
        if (t < s) red[t] += red[t + s];
        __syncthreads();
    }
    if (t == 0) {
        int len = red[0];
        int br  = branch_idx[b];
        int w   = 128 << br;          // 128, 256, 512
        int Si  = T_LEN / w;          // 512, 256, 128
        int nv  = len / w;
        if (nv > Si) nv = Si;
        nv_sh = nv;
        n_valid_ws[b] = nv;
    }
    __syncthreads();
    const int nv = nv_sh;
    for (int s = t; s < S_MAX; s += 256)
        mask_out[(size_t)b * S_MAX + s] = (s < nv) ? 1.0f : 0.0f;
}

// ---------------- Kernel 2: per-sample selected-branch GEMM via f32 WMMA ----
// Double-buffered LDS, async global->LDS staging overlapped with WMMA compute.
__global__ __launch_bounds__(256)
void branch_gemm_kernel(const float* __restrict__ signal,
                        const int*   __restrict__ branch_idx,
                        const float* __restrict__ W0, const float* __restrict__ bias0,
                        const float* __restrict__ W1, const float* __restrict__ bias1,
                        const float* __restrict__ W2, const float* __restrict__ bias2,
                        const int*   __restrict__ n_valid_ws,
                        float* __restrict__ out) {
    __shared__ float lds_a[2][MT * LDP];   // 2 x (64 x 32 +pad) patch tiles
    __shared__ float lds_b[2][NT * LDP];   // 2 x (128 x 32 +pad) weight slices

    const int b      = blockIdx.z;
    const int m_base = blockIdx.y * MT;
    const int n_base = blockIdx.x * NT;
    const int t      = threadIdx.x;
    const int lane   = t & 31;
    const int wave   = t >> 5;
    const int wave_m = wave >> 1;       // 0..3 -> 16-row slice
    const int wave_n = wave & 1;        // 0..1 -> 64-col slice

    const int br = branch_idx[b];
    const int w  = 128 << br;
    const int Si = T_LEN / w;

    float* out_tile = out + (size_t)b * S_MAX * E_DIM;

    if (m_base >= Si) {
        // Padded region beyond this sample's token count: zero-fill tile.
        const float4 z = make_float4(0.f, 0.f, 0.f, 0.f);
        for (int i = t; i < MT * NT / 4; i += 256) {
            int r = (i * 4) / NT;
            int c = (i * 4) % NT;
            *(float4*)(out_tile + (size_t)(m_base + r) * E_DIM + n_base + c) = z;
        }
        return;
    }

    const int nv = n_valid_ws[b];
    const float* Wsel = (br == 0) ? W0 : (br == 1) ? W1 : W2;
    const float* bsel = (br == 0) ? bias0 : (br == 1) ? bias1 : bias2;
    const float* sig  = signal + (size_t)b * T_LEN;

    // Per-thread staging coordinates: 16B per lane per async op.
    const int rowA = t >> 3;            // 0..31
    const int c4   = (t & 7) << 2;      // 0,4,...,28
    const float* gA0 = sig  + (size_t)(m_base + rowA)      * w + c4;
    const float* gA1 = sig  + (size_t)(m_base + rowA + 32) * w + c4;
    const float* gB0 = Wsel + (size_t)(n_base + rowA)      * w + c4;
    const float* gB1 = Wsel + (size_t)(n_base + rowA + 32) * w + c4;
    const float* gB2 = Wsel + (size_t)(n_base + rowA + 64) * w + c4;
    const float* gB3 = Wsel + (size_t)(n_base + rowA + 96) * w + c4;

    auto stage = [&](int buf, int kc) {
        async_cp16(gA0 + kc, &lds_a[buf][rowA * LDP + c4]);
        async_cp16(gA1 + kc, &lds_a[buf][(rowA + 32) * LDP + c4]);
        async_cp16(gB0 + kc, &lds_b[buf][rowA * LDP + c4]);
        async_cp16(gB1 + kc, &lds_b[buf][(rowA + 32) * LDP + c4]);
        async_cp16(gB2 + kc, &lds_b[buf][(rowA + 64) * LDP + c4]);
        async_cp16(gB3 + kc, &lds_b[buf][(rowA + 96) * LDP + c4]);
    };

    v8f zero;
    #pragma unroll
    for (int j = 0; j < 8; ++j) zero[j] = 0.0f;
    v8f acc[4];
    #pragma unroll
    for (int i = 0; i < 4; ++i) acc[i] = zero;

    const int koff = (lane >> 4) << 1;  // lane half selects K pair {0,1} / {2,3}
    const int nloc = lane & 15;

    // Prologue: fill buffer 0.
    stage(0, 0);
    wait_async0();
    __syncthreads();

    int buf = 0;
    for (int kc = 0; kc < w; kc += KC) {
        // Prefetch next chunk into the other buffer (background DMA).
        if (kc + KC < w) stage(buf ^ 1, kc + KC);

        const float* arow = &lds_a[buf][(wave_m * 16 + nloc) * LDP];
        #pragma unroll
        for (int kk = 0; kk < KC; kk += 4) {
            v2f a;
            a.x = arow[kk + koff];
            a.y = arow[kk + koff + 1];
            #pragma unroll
            for (int nt = 0; nt < 4; ++nt) {
                const float* brow = &lds_b[buf][(wave_n * 64 + nt * 16 + nloc) * LDP];
                v2f bf;
                bf.x = brow[kk + koff];
                bf.y = brow[kk + koff + 1];
                acc[nt] = __builtin_amdgcn_wmma_f32_16x16x4_f32(
                    false, a, false, bf, (short)0, acc[nt], false, false);
            }
        }

        wait_async0();     // next buffer fully landed in LDS
        __syncthreads();   // all waves done reading current buffer
        buf ^= 1;
    }

    // Epilogue: bias + n_valid row mask + store
    const int m_half = (lane >> 4) * 8;  // C layout: VGPR r -> row r (lo) / r+8 (hi)
    #pragma unroll
    for (int nt = 0; nt < 4; ++nt) {
        int e = n_base + wave_n * 64 + nt * 16 + nloc;
        float bv = bsel[e];
        #pragma unroll
        for (int r = 0; r < 8; ++r) {
            int m = m_base + wave_m * 16 + m_half + r;
            float v = acc[nt][r] + bv;
            out_tile[(size_t)m * E_DIM + e] = (m < nv) ? v : 0.0f;
        }
    }
}

// ---------------------------------------------------------------------------
extern "C" void kernel_launch(void* const* d_in, const int* in_sizes, int n_in,
                              void* d_out, int out_size, void* d_ws, size_t ws_size,
                              hipStream_t stream) {
    const float*         signal      = (const float*)d_in[0];
    const unsigned char* signal_mask = (const unsigned char*)d_in[1];
    const int*           branch_idx  = (const int*)d_in[2];
    const float*         W0 = (const float*)d_in[3];
    const float*         b0 = (const float*)d_in[4];
    const float*         W1 = (const float*)d_in[5];
    const float*         b1 = (const float*)d_in[6];
    const float*         W2 = (const float*)d_in[7];
    const float*         b2 = (const float*)d_in[8];

    float* tokens_out = (float*)d_out;
    float* mask_out   = tokens_out + (size_t)B_N * S_MAX * E_DIM;
    int*   n_valid_ws = (int*)d_ws;

    mask_reduce_kernel<<<B_N, 256, 0, stream>>>(signal_mask, branch_idx,
                                                mask_out, n_valid_ws);

    dim3 grid(E_DIM / NT, S_MAX / MT, B_N);   // (4, 8, 128)
    branch_gemm_kernel<<<grid, 256, 0, stream>>>(signal, branch_idx,
                                                 W0, b0, W1, b1, W2, b2,
                                                 n_valid_ws, tokens_out);
}